// Codebook_89799176224925
// MI455X (gfx1250) — compile-verified
//
#include <hip/hip_runtime.h>
#include <hip/hip_bf16.h>

// VQ-VAE codebook quantization for MI455X (gfx1250, wave32).
// N=16384 rows, D=256 latent, K=8192 codes. f32 WMMA (16x16x4) for the
// distance GEMM: emb values are ~1e-4 so bf16/f16 would corrupt the argmin.
// Each wave multiplies every B (codebook) fragment against TWO 16-row A
// tiles -> halves L2 emb traffic and doubles WMMA density per load.

#define NCODES 8192
#define LATD   256
#define NROWS  16384
#define ROWS_PER_BLOCK 32
#define OUT_ZQ_ELEMS (NROWS * LATD)      // 4194304
#define GATHER_BLOCKS 2048               // 2048 blocks * 2048 elems = 4194304

typedef __attribute__((ext_vector_type(2))) float v2f;
typedef __attribute__((ext_vector_type(8))) float v8f;

// ---------------------------------------------------------------- norms ----
// One wave per code: coalesced read of one 256-float row, shfl-xor reduce.
__global__ __launch_bounds__(256)
void vq_norms(const float* __restrict__ emb, float* __restrict__ norms) {
    const int code = (blockIdx.x * blockDim.x + threadIdx.x) >> 5;
    const int lane = threadIdx.x & 31;
    const float* row = emb + (size_t)code * LATD;
    float s = 0.0f;
    #pragma unroll
    for (int d = 0; d < LATD / 32; ++d) {
        float v = row[lane + d * 32];
        s = fmaf(v, v, s);
    }
    #pragma unroll
    for (int m = 16; m >= 1; m >>= 1) s += __shfl_xor(s, m, 32);
    if (lane == 0) norms[code] = s;
}

// --------------------------------------------------------------- argmin ----
// Block = 4 waves, owns 32 rows of z (LDS tile). Each wave scans 2048 codes
// in 16x64 tiles; every B fragment feeds two M-tiles (8 WMMAs / K-step).
// dist = ||e||^2 - 2*S  (||z||^2 is row-constant: dropped from the argmin).
__global__ __launch_bounds__(128)
void vq_argmin(const float* __restrict__ z, const float* __restrict__ emb,
               const float* __restrict__ norms, int* __restrict__ out_idx) {
    __shared__ float zt[ROWS_PER_BLOCK * LATD];   // 32 KB z tile
    __shared__ float red_val[4][ROWS_PER_BLOCK];
    __shared__ int   red_idx[4][ROWS_PER_BLOCK];

    const int tid    = threadIdx.x;
    const int waveId = tid >> 5;
    const int lane   = tid & 31;
    const int half   = lane >> 4;        // A/B VGPR layout: k = kb + 2*half + j
    const int lanelo = lane & 15;        // A: m index, B: n index, C/D: n index
    const size_t rowbase = (size_t)blockIdx.x * ROWS_PER_BLOCK;

    // Stage the 32x256 z tile into LDS (coalesced, all 128 threads).
    for (int i = tid; i < ROWS_PER_BLOCK * LATD; i += 128)
        zt[i] = z[rowbase * LATD + i];
    __syncthreads();

    const int codeBase = waveId * (NCODES / 4);   // 2048 codes per wave

    float minv0[8], minv1[8];
    int   mini0[8], mini1[8];
    #pragma unroll
    for (int r = 0; r < 8; ++r) {
        minv0[r] = 3.4e38f; mini0[r] = 0;
        minv1[r] = 3.4e38f; mini1[r] = 0;
    }

    const float* ap0 = &zt[lanelo * LATD + half * 2];          // rows 0..15
    const float* ap1 = &zt[(lanelo + 16) * LATD + half * 2];   // rows 16..31

    for (int t = 0; t < NCODES / 4; t += 64) {
        const int c0 = codeBase + t + lanelo;     // n for sub-tile 0 (this lane)
        const float* bp0 = emb + (size_t)(c0     ) * LATD + half * 2;
        const float* bp1 = emb + (size_t)(c0 + 16) * LATD + half * 2;
        const float* bp2 = emb + (size_t)(c0 + 32) * LATD + half * 2;
        const float* bp3 = emb + (size_t)(c0 + 48) * LATD + half * 2;

        v8f a00 = {}, a01 = {}, a02 = {}, a03 = {};   // M-tile 0 accumulators
        v8f a10 = {}, a11 = {}, a12 = {}, a13 = {};   // M-tile 1 accumulators

        #pragma unroll 4
        for (int kb = 0; kb < LATD; kb += 4) {
            v2f f0 = *(const v2f*)(ap0 + kb);     // ds_load_b64 (A tile 0)
            v2f f1 = *(const v2f*)(ap1 + kb);     // ds_load_b64 (A tile 1)
            v2f b0 = *(const v2f*)(bp0 + kb);     // global_load_b64 (L2-resident)
            v2f b1 = *(const v2f*)(bp1 + kb);
            v2f b2 = *(const v2f*)(bp2 + kb);
            v2f b3 = *(const v2f*)(bp3 + kb);
            a00 = __builtin_amdgcn_wmma_f32_16x16x4_f32(false, f0, false, b0,
                                                        (short)0, a00, false, false);
            a10 = __builtin_amdgcn_wmma_f32_16x16x4_f32(false, f1, false, b0,
                                                        (short)0, a10, false, false);
            a01 = __builtin_amdgcn_wmma_f32_16x16x4_f32(false, f0, false, b1,
                                                        (short)0, a01, false, false);
            a11 = __builtin_amdgcn_wmma_f32_16x16x4_f32(false, f1, false, b1,
                                                        (short)0, a11, false, false);
            a02 = __builtin_amdgcn_wmma_f32_16x16x4_f32(false, f0, false, b2,
                                                        (short)0, a02, false, false);
            a12 = __builtin_amdgcn_wmma_f32_16x16x4_f32(false, f1, false, b2,
                                                        (short)0, a12, false, false);
            a03 = __builtin_amdgcn_wmma_f32_16x16x4_f32(false, f0, false, b3,
                                                        (short)0, a03, false, false);
            a13 = __builtin_amdgcn_wmma_f32_16x16x4_f32(false, f1, false, b3,
                                                        (short)0, a13, false, false);
        }

        const float n0 = norms[c0];
        const float n1 = norms[c0 + 16];
        const float n2 = norms[c0 + 32];
        const float n3 = norms[c0 + 48];
        #pragma unroll
        for (int r = 0; r < 8; ++r) {
            float d;
            d = fmaf(-2.0f, a00[r], n0);
            if (d < minv0[r]) { minv0[r] = d; mini0[r] = c0; }
            d = fmaf(-2.0f, a01[r], n1);
            if (d < minv0[r]) { minv0[r] = d; mini0[r] = c0 + 16; }
            d = fmaf(-2.0f, a02[r], n2);
            if (d < minv0[r]) { minv0[r] = d; mini0[r] = c0 + 32; }
            d = fmaf(-2.0f, a03[r], n3);
            if (d < minv0[r]) { minv0[r] = d; mini0[r] = c0 + 48; }

            d = fmaf(-2.0f, a10[r], n0);
            if (d < minv1[r]) { minv1[r] = d; mini1[r] = c0; }
            d = fmaf(-2.0f, a11[r], n1);
            if (d < minv1[r]) { minv1[r] = d; mini1[r] = c0 + 16; }
            d = fmaf(-2.0f, a12[r], n2);
            if (d < minv1[r]) { minv1[r] = d; mini1[r] = c0 + 32; }
            d = fmaf(-2.0f, a13[r], n3);
            if (d < minv1[r]) { minv1[r] = d; mini1[r] = c0 + 48; }
        }
    }

    // Per-row reduce: row m = r + 8*half lives across the 16 lanes of a half.
    // xor masks 1,2,4,8 stay inside each 16-lane half. Tie-break: lowest idx.
    #pragma unroll
    for (int r = 0; r < 8; ++r) {
        float bv = minv0[r];
        int   bi = mini0[r];
        #pragma unroll
        for (int m = 1; m < 16; m <<= 1) {
            float ov = __shfl_xor(bv, m, 32);
            int   oi = __shfl_xor(bi, m, 32);
            if (ov < bv || (ov == bv && oi < bi)) { bv = ov; bi = oi; }
        }
        minv0[r] = bv; mini0[r] = bi;

        bv = minv1[r];
        bi = mini1[r];
        #pragma unroll
        for (int m = 1; m < 16; m <<= 1) {
            float ov = __shfl_xor(bv, m, 32);
            int   oi = __shfl_xor(bi, m, 32);
            if (ov < bv || (ov == bv && oi < bi)) { bv = ov; bi = oi; }
        }
        minv1[r] = bv; mini1[r] = bi;
    }
    if (lanelo == 0) {
        #pragma unroll
        for (int r = 0; r < 8; ++r) {
            const int m = r + 8 * half;
            red_val[waveId][m]      = minv0[r];
            red_idx[waveId][m]      = mini0[r];
            red_val[waveId][m + 16] = minv1[r];
            red_idx[waveId][m + 16] = mini1[r];
        }
    }
    __syncthreads();

    // Cross-wave combine (waves own ascending disjoint code ranges).
    if (tid < ROWS_PER_BLOCK) {
        float bv = red_val[0][tid];
        int   bi = red_idx[0][tid];
        #pragma unroll
        for (int w = 1; w < 4; ++w) {
            float ov = red_val[w][tid];
            int   oi = red_idx[w][tid];
            if (ov < bv || (ov == bv && oi < bi)) { bv = ov; bi = oi; }
        }
        out_idx[rowbase + tid] = bi;
    }
}

// --------------------------------------------------------------- gather ----
// z_q_st = z + (z_q - z) (bit-matching the straight-through expression),
// plus deterministic per-block partial sums of (z_q - z)^2.
__global__ __launch_bounds__(256)
void vq_gather(const float* __restrict__ z, const float* __restrict__ emb,
               const int* __restrict__ idxs, float* __restrict__ out_zq,
               float* __restrict__ partial) {
    __shared__ float sred[256];
    const int tid = threadIdx.x;
    const size_t base = (size_t)blockIdx.x * 2048;
    float s = 0.0f;
    #pragma unroll
    for (int j = 0; j < 8; ++j) {
        const size_t e   = base + (size_t)j * 256 + tid;
        const int    row = (int)(e >> 8);
        const int    d   = (int)(e & 255);
        const float  q   = emb[(size_t)idxs[row] * LATD + d];
        const float  zz  = z[e];
        const float  df  = q - zz;
        out_zq[e] = zz + df;                 // z + sg(z_q - z)
        s = fmaf(df, df, s);
    }
    sred[tid] = s;
    __syncthreads();
    for (int step = 128; step > 0; step >>= 1) {
        if (tid < step) sred[tid] += sred[tid + step];
        __syncthreads();
    }
    if (tid == 0) partial[blockIdx.x] = sred[0];
}

// ---------------------------------------------------------------- final ----
// Single block: deterministic sum of partials -> loss; idx -> float output.
__global__ __launch_bounds__(256)
void vq_final(const float* __restrict__ partial, const int* __restrict__ idxs,
              float* __restrict__ out) {
    __shared__ float sred[256];
    const int tid = threadIdx.x;
    float s = 0.0f;
    for (int i = tid; i < GATHER_BLOCKS; i += 256) s += partial[i];
    sred[tid] = s;
    __syncthreads();
    for (int step = 128; step > 0; step >>= 1) {
        if (tid < step) sred[tid] += sred[tid + step];
        __syncthreads();
    }
    for (int i = tid; i < NROWS; i += 256)
        out[OUT_ZQ_ELEMS + i] = (float)idxs[i];
    if (tid == 0)
        out[OUT_ZQ_ELEMS + NROWS] =
            sred[0] * ((1.0f + 0.25f) / (float)OUT_ZQ_ELEMS);
}

// --------------------------------------------------------------- launch ----
extern "C" void kernel_launch(void* const* d_in, const int* in_sizes, int n_in,
                              void* d_out, int out_size, void* d_ws, size_t ws_size,
                              hipStream_t stream) {
    (void)in_sizes; (void)n_in; (void)out_size; (void)ws_size;
    const float* z   = (const float*)d_in[0];   // [16,1024,256] f32
    const float* emb = (const float*)d_in[1];   // [8192,256] f32
    float* out = (float*)d_out;                 // [zq | idx-as-f32 | loss]

    char* ws = (char*)d_ws;
    int*   ws_idx     = (int*)ws;                           // 64 KB
    float* ws_partial = (float*)(ws + 65536);               //  8 KB
    float* ws_norms   = (float*)(ws + 65536 + 8192);        // 32 KB

    vq_norms <<<NCODES * 32 / 256,    256, 0, stream>>>(emb, ws_norms);
    vq_argmin<<<NROWS / ROWS_PER_BLOCK, 128, 0, stream>>>(z, emb, ws_norms, ws_idx);
    vq_gather<<<GATHER_BLOCKS,        256, 0, stream>>>(z, emb, ws_idx, out, ws_partial);
    vq_final <<<1,                    256, 0, stream>>>(ws_partial, ws_idx, out);
}